// LocalFWLNet_12816182411670
// MI455X (gfx1250) — compile-verified
//
#include <hip/hip_runtime.h>

#define EPSC 1e-5f

typedef __attribute__((ext_vector_type(2))) float v2f;
typedef __attribute__((ext_vector_type(8))) float v8f;

// ---------------------------------------------------------------------------
// WMMA helpers: V_WMMA_F32_16X16X4_F32, fragment layouts per CDNA5 ISA 7.12.2
//   A (16x4 f32): lanes 0-15 hold (M=lane, K=k0,k0+1); lanes 16-31 hold K=k0+2,k0+3
//   B (4x16 f32): lanes hold N=lane%16; VGPR pair holds K rows split like A
//   C/D (16x16 f32): VGPR r, lanes0-15 -> M=r, lanes16-31 -> M=r+8
// ---------------------------------------------------------------------------
__device__ __forceinline__ v8f wmma4(v2f a, v2f b, v8f c) {
  return __builtin_amdgcn_wmma_f32_16x16x4_f32(false, a, false, b, (short)0, c,
                                               false, false);
}

// A fragment from a precomputed (clamped) row pointer; msk zeroes OOB rows.
// Unconditional loads -> no EXEC save/restore in the K-loop.
__device__ __forceinline__ v2f load_a_frag(const float* arow, float msk, int k0) {
  int lane = threadIdx.x & 31;
  int kb = k0 + ((lane >> 4) << 1);
  v2f a;
  a.x = arow[kb] * msk;
  a.y = arow[kb + 1] * msk;
  return a;
}

__device__ __forceinline__ v2f load_b32(const float* B, int k0, int n0) {
  int lane = threadIdx.x & 31;
  int n = n0 + (lane & 15);
  int kb = k0 + ((lane >> 4) << 1);
  v2f b;
  b.x = B[kb * 32 + n];
  b.y = B[(kb + 1) * 32 + n];
  return b;
}

// C/D store with wave-uniform fast path: only the last stripe is partial.
__device__ __forceinline__ void store_c(float* D, int row0, int M, int n0, v8f c) {
  int lane = threadIdx.x & 31;
  int n = n0 + (lane & 15);
  int mo = (lane >> 4) << 3;
  if (row0 + 16 <= M) {  // uniform: straight-line stores, no EXEC churn
#pragma unroll
    for (int r = 0; r < 8; ++r) D[(row0 + r + mo) * 32 + n] = c[r];
  } else {
#pragma unroll
    for (int r = 0; r < 8; ++r) {
      int m = row0 + r + mo;
      if (m < M) D[m * 32 + n] = c[r];
    }
  }
}

__device__ __forceinline__ void store_c_bias_relu(float* D, int row0, int M,
                                                  int n0, v8f c,
                                                  const float* bias) {
  int lane = threadIdx.x & 31;
  int n = n0 + (lane & 15);
  float bv = bias[n];
  int mo = (lane >> 4) << 3;
  if (row0 + 16 <= M) {  // uniform fast path
#pragma unroll
    for (int r = 0; r < 8; ++r) {
      float v = c[r] + bv;
      D[(row0 + r + mo) * 32 + n] = v > 0.0f ? v : 0.0f;
    }
  } else {
#pragma unroll
    for (int r = 0; r < 8; ++r) {
      int m = row0 + r + mo;
      if (m < M) {
        float v = c[r] + bv;
        D[m * 32 + n] = v > 0.0f ? v : 0.0f;
      }
    }
  }
}

// ---------------------------------------------------------------------------
// Small utility kernels
// ---------------------------------------------------------------------------
__global__ void k_zero(float* p, long long n) {
  long long i = (long long)blockIdx.x * blockDim.x + threadIdx.x;
  if (i < n) p[i] = 0.0f;
}

__global__ void k_init_deg(float* deg, int n) {
  int i = blockIdx.x * blockDim.x + threadIdx.x;
  if (i < n) deg[i] = 1.0f;  // +1 self loop
}

__global__ void k_add_deg(const int* dst, float* deg, int m) {
  int e = blockIdx.x * blockDim.x + threadIdx.x;
  if (e < m) atomicAdd(&deg[dst[e]], 1.0f);
}

__global__ void k_dinv(const float* deg, float* dinv, int n) {
  int i = blockIdx.x * blockDim.x + threadIdx.x;
  if (i < n) dinv[i] = rsqrtf(deg[i]);
}

__global__ void k_gather_emb(const float* emb, const int* xn, float* x, int n) {
  int idx = blockIdx.x * blockDim.x + threadIdx.x;
  if (idx >= n * 32) return;
  int i = idx >> 5, c = idx & 31;
  x[idx] = emb[xn[i] * 32 + c];
}

// ---------------------------------------------------------------------------
// GEMM: D[M,32] = A[M,32] @ W[32,32]   (one wave per 16-row stripe, 2 N-tiles)
// ---------------------------------------------------------------------------
__global__ void k_gemm_k32(const float* __restrict__ A,
                           const float* __restrict__ W, float* __restrict__ D,
                           int M) {
  int wid = blockIdx.x * (blockDim.x >> 5) + (threadIdx.x >> 5);
  int row0 = wid * 16;
  if (row0 >= M) return;  // wave-uniform: EXEC stays all-1 for WMMA
  int lane = threadIdx.x & 31;
  int m = row0 + (lane & 15);
  float msk = (m < M) ? 1.0f : 0.0f;
  const float* arow = A + (size_t)(m < M ? m : M - 1) * 32;
  v8f c0 = {}, c1 = {};
#pragma unroll
  for (int k0 = 0; k0 < 32; k0 += 4) {
    v2f a = load_a_frag(arow, msk, k0);
    v2f b0 = load_b32(W, k0, 0);
    v2f b1 = load_b32(W, k0, 16);
    c0 = wmma4(a, b0, c0);
    c1 = wmma4(a, b1, c1);
  }
  store_c(D, row0, M, 0, c0);
  store_c(D, row0, M, 16, c1);
}

// GCN self-loop + bias init: acc = h / deg + bias
__global__ void k_gcn_init(const float* h, const float* deg, const float* bias,
                           float* acc, int n) {
  int idx = blockIdx.x * blockDim.x + threadIdx.x;
  if (idx >= n * 32) return;
  int i = idx >> 5, c = idx & 31;
  acc[idx] = h[idx] / deg[i] + bias[c];
}

// GCN message scatter: acc[dst] += h[src] * dinv[src]*dinv[dst]
__global__ void k_gcn_scatter(const float* __restrict__ h,
                              const int* __restrict__ src,
                              const int* __restrict__ dst,
                              const float* __restrict__ dinv, float* acc,
                              int m) {
  int idx = blockIdx.x * blockDim.x + threadIdx.x;
  if (idx >= m * 32) return;
  int e = idx >> 5, c = idx & 31;
  int s = src[e], d = dst[e];
  atomicAdd(&acc[d * 32 + c], h[s * 32 + c] * dinv[s] * dinv[d]);
}

// Column sum & sum-of-squares reduction (stats[0:32]=sum, stats[32:64]=sumsq)
__global__ void k_colstats(const float* __restrict__ X, int M, float* stats) {
  int c = threadIdx.x & 31;
  int w = blockIdx.x * (blockDim.x >> 5) + (threadIdx.x >> 5);
  int stride = gridDim.x * (blockDim.x >> 5);
  float s = 0.0f, ss = 0.0f;
  for (int r = w; r < M; r += stride) {
    float v = X[r * 32 + c];
    s += v;
    ss += v * v;
  }
  atomicAdd(&stats[c], s);
  atomicAdd(&stats[32 + c], ss);
}

// GraphNorm + ReLU.  var = E[h^2] - (2*ms - ms^2)*mean^2
__global__ void k_gnorm_relu(const float* __restrict__ Z, float* __restrict__ Y,
                             int M, const float* __restrict__ stats,
                             const float* gw, const float* gb,
                             const float* gms) {
  int idx = blockIdx.x * blockDim.x + threadIdx.x;
  if (idx >= M * 32) return;
  int c = idx & 31;
  float n = (float)M;
  float mean = stats[c] / n;
  float ms = gms[c];
  float var = stats[32 + c] / n - (2.0f * ms - ms * ms) * mean * mean;
  float out = Z[idx] - ms * mean;
  float v = gw[c] * out * rsqrtf(var + EPSC) + gb[c];
  Y[idx] = v > 0.0f ? v : 0.0f;
}

// ---------------------------------------------------------------------------
// Edge MLPs: val[e] = concat(x[src[e]], x[dst[e]]) (gathered on the fly)
//   xe  = relu(val @ mlp1_w + b1)
//   mul = relu(val @ mlp2_w + b2)
// One wave per 16-edge stripe; A fragment reused across 4 B tiles per k-step.
// ---------------------------------------------------------------------------
__global__ void k_edge_mlp(const float* __restrict__ x,
                           const int* __restrict__ src,
                           const int* __restrict__ dst,
                           const float* __restrict__ w1, const float* b1v,
                           const float* __restrict__ w2, const float* b2v,
                           float* xe, float* mul, int m) {
  int wid = blockIdx.x * (blockDim.x >> 5) + (threadIdx.x >> 5);
  int row0 = wid * 16;
  if (row0 >= m) return;  // wave-uniform
  int lane = threadIdx.x & 31;
  int e = row0 + (lane & 15);
  int ec = e < m ? e : m - 1;
  float msk = (e < m) ? 1.0f : 0.0f;
  const float* rs = x + (size_t)src[ec] * 32;
  const float* rd = x + (size_t)dst[ec] * 32;
  v8f c10 = {}, c11 = {}, c20 = {}, c21 = {};
#pragma unroll
  for (int k0 = 0; k0 < 64; k0 += 4) {
    int kb = k0 + ((lane >> 4) << 1);
    v2f a;
    a.x = ((kb < 32) ? rs[kb] : rd[kb - 32]) * msk;
    a.y = ((kb + 1 < 32) ? rs[kb + 1] : rd[kb + 1 - 32]) * msk;
    v2f p0 = load_b32(w1, k0, 0);
    v2f p1 = load_b32(w1, k0, 16);
    v2f q0 = load_b32(w2, k0, 0);
    v2f q1 = load_b32(w2, k0, 16);
    c10 = wmma4(a, p0, c10);
    c11 = wmma4(a, p1, c11);
    c20 = wmma4(a, q0, c20);
    c21 = wmma4(a, q1, c21);
  }
  store_c_bias_relu(xe, row0, m, 0, c10, b1v);
  store_c_bias_relu(xe, row0, m, 16, c11, b1v);
  store_c_bias_relu(mul, row0, m, 0, c20, b2v);
  store_c_bias_relu(mul, row0, m, 16, c21, b2v);
}

// sparse-sparse bmm: bmm[tri_seg[t]] += xe[tri_a[t]] * mul[tri_b[t]]
__global__ void k_bmm_scatter(const float* __restrict__ xe,
                              const float* __restrict__ mul,
                              const int* __restrict__ ta,
                              const int* __restrict__ tb,
                              const int* __restrict__ ts, float* bmm,
                              long long total) {
  long long idx = (long long)blockIdx.x * blockDim.x + threadIdx.x;
  if (idx >= total * 32) return;
  long long t = idx >> 5;
  int c = (int)(idx & 31);
  int a = ta[t], b = tb[t], s = ts[t];
  atomicAdd(&bmm[(long long)s * 32 + c], xe[(long long)a * 32 + c] *
                                             mul[(long long)b * 32 + c]);
}

// mlp3: z[M,32] = bmm[M,32] @ w3[0:32,:] + is_edge[m]*w3[32,:] + b3
__global__ void k_mlp3(const float* __restrict__ A,
                       const float* __restrict__ W3,  // (33,32) row-major
                       const float* __restrict__ is_edge, const float* b3,
                       float* __restrict__ Z, int M) {
  int wid = blockIdx.x * (blockDim.x >> 5) + (threadIdx.x >> 5);
  int row0 = wid * 16;
  if (row0 >= M) return;
  int lane = threadIdx.x & 31;
  int mrow = row0 + (lane & 15);
  float msk = (mrow < M) ? 1.0f : 0.0f;
  const float* arow = A + (size_t)(mrow < M ? mrow : M - 1) * 32;
  v8f c0 = {}, c1 = {};
#pragma unroll
  for (int k0 = 0; k0 < 32; k0 += 4) {
    v2f a = load_a_frag(arow, msk, k0);
    v2f b0 = load_b32(W3, k0, 0);
    v2f b1 = load_b32(W3, k0, 16);
    c0 = wmma4(a, b0, c0);
    c1 = wmma4(a, b1, c1);
  }
  const float* w3last = W3 + 32 * 32;  // row 32: is_edge channel weights
  int n = lane & 15;
  int mo = (lane >> 4) << 3;
  if (row0 + 16 <= M) {  // uniform fast path
#pragma unroll
    for (int r = 0; r < 8; ++r) {
      int m = row0 + r + mo;
      float ie = is_edge[m];
      Z[m * 32 + n] = c0[r] + ie * w3last[n] + b3[n];
      Z[m * 32 + n + 16] = c1[r] + ie * w3last[n + 16] + b3[n + 16];
    }
  } else {
#pragma unroll
    for (int r = 0; r < 8; ++r) {
      int m = row0 + r + mo;
      if (m < M) {
        float ie = is_edge[m];
        Z[m * 32 + n] = c0[r] + ie * w3last[n] + b3[n];
        Z[m * 32 + n + 16] = c1[r] + ie * w3last[n + 16] + b3[n + 16];
      }
    }
  }
}

// xo(i,j) * xo(j,i) via transpose permutation
__global__ void k_transmul(const float* __restrict__ xo,
                           const int* __restrict__ perm, float* y,
                           long long n_out) {
  long long idx = (long long)blockIdx.x * blockDim.x + threadIdx.x;
  if (idx >= n_out * 32) return;
  long long i = idx >> 5;
  int c = (int)(idx & 31);
  y[idx] = xo[idx] * xo[(long long)perm[i] * 32 + c];
}

// final readout: out[p] = dot(concat(y[pred_idx]*mask, x[p0]*x[p1]), lind_w)+b
__global__ void k_final(const float* __restrict__ y,
                        const float* __restrict__ x,
                        const int* __restrict__ pos,
                        const int* __restrict__ pred_idx,
                        const float* __restrict__ pred_mask,
                        const float* __restrict__ lw, const float* lb,
                        float* out, int P) {
  int p = blockIdx.x * blockDim.x + threadIdx.x;
  if (p >= P) return;
  float acc = lb[0];
  long long pi = pred_idx[p];
  float pm = pred_mask[p];
  const float* yr = y + pi * 32;
#pragma unroll
  for (int c = 0; c < 32; ++c) acc += yr[c] * pm * lw[c];
  const float* x0 = x + (long long)pos[2 * p] * 32;
  const float* x1 = x + (long long)pos[2 * p + 1] * 32;
#pragma unroll
  for (int c = 0; c < 32; ++c) acc += x0[c] * x1[c] * lw[32 + c];
  out[p] = acc;
}

// ---------------------------------------------------------------------------
// Host orchestration
// ---------------------------------------------------------------------------
static inline int cdiv(long long a, long long b) { return (int)((a + b - 1) / b); }

extern "C" void kernel_launch(void* const* d_in, const int* in_sizes, int n_in,
                              void* d_out, int out_size, void* d_ws,
                              size_t ws_size, hipStream_t stream) {
  // inputs (setup_inputs dict order)
  const float* emb = (const float*)d_in[0];
  const float* gcn1_w = (const float*)d_in[1];
  const float* gcn1_b = (const float*)d_in[2];
  const float* gn1_w = (const float*)d_in[3];
  const float* gn1_b = (const float*)d_in[4];
  const float* gn1_ms = (const float*)d_in[5];
  const float* gcn2_w = (const float*)d_in[6];
  const float* gcn2_b = (const float*)d_in[7];
  const float* gn2_w = (const float*)d_in[8];
  const float* gn2_b = (const float*)d_in[9];
  const float* gn2_ms = (const float*)d_in[10];
  const float* mlp1_w = (const float*)d_in[11];
  const float* mlp1_b = (const float*)d_in[12];
  const float* mlp2_w = (const float*)d_in[13];
  const float* mlp2_b = (const float*)d_in[14];
  const float* mlp3_w = (const float*)d_in[15];
  const float* mlp3_b = (const float*)d_in[16];
  const float* gn3_w = (const float*)d_in[17];
  const float* gn3_b = (const float*)d_in[18];
  const float* gn3_ms = (const float*)d_in[19];
  const float* lind_w = (const float*)d_in[20];
  const float* lind_b = (const float*)d_in[21];
  const float* is_edge = (const float*)d_in[22];
  const float* pred_mask = (const float*)d_in[23];
  const int* x_nodes = (const int*)d_in[24];
  const int* ei = (const int*)d_in[25];
  const int* pos = (const int*)d_in[26];
  const int* tri_a = (const int*)d_in[27];
  const int* tri_b = (const int*)d_in[28];
  const int* tri_seg = (const int*)d_in[29];
  const int* trans_perm = (const int*)d_in[30];
  const int* pred_idx = (const int*)d_in[31];

  const int N = in_sizes[24];
  const int m = in_sizes[25] / 2;
  const long long n_out = in_sizes[22];
  const long long total = in_sizes[27];
  const int P = in_sizes[31];

  const int* src = ei;      // ei[0]
  const int* dst = ei + m;  // ei[1]

  // workspace layout (floats)
  float* ws = (float*)d_ws;
  float* deg = ws;                      // N
  float* dinv = deg + N;                // N
  float* xA = dinv + N;                 // N*32  (current node feats)
  float* xB = xA + (size_t)N * 32;      // N*32  (h = x @ W)
  float* xC = xB + (size_t)N * 32;      // N*32  (aggregation)
  float* stats = xC + (size_t)N * 32;   // 64
  float* xe = stats + 64;               // m*32
  float* mul = xe + (size_t)m * 32;     // m*32
  float* bmm = mul + (size_t)m * 32;    // n_out*32
  float* z = bmm + n_out * 32;          // n_out*32 (pre-norm, then xo in place)
  float* y = z + n_out * 32;            // n_out*32

  const int T = 256;

  // ---- degree / norm ----
  k_init_deg<<<cdiv(N, T), T, 0, stream>>>(deg, N);
  k_add_deg<<<cdiv(m, T), T, 0, stream>>>(dst, deg, m);
  k_dinv<<<cdiv(N, T), T, 0, stream>>>(deg, dinv, N);

  // ---- embedding gather ----
  k_gather_emb<<<cdiv((long long)N * 32, T), T, 0, stream>>>(emb, x_nodes, xA, N);

  const int gemmBlocks = cdiv(cdiv(N, 16), T / 32);

  // ---- GCN layer 1 ----
  k_gemm_k32<<<gemmBlocks, T, 0, stream>>>(xA, gcn1_w, xB, N);
  k_gcn_init<<<cdiv((long long)N * 32, T), T, 0, stream>>>(xB, deg, gcn1_b, xC, N);
  k_gcn_scatter<<<cdiv((long long)m * 32, T), T, 0, stream>>>(xB, src, dst, dinv, xC, m);
  k_zero<<<1, 64, 0, stream>>>(stats, 64);
  k_colstats<<<256, T, 0, stream>>>(xC, N, stats);
  k_gnorm_relu<<<cdiv((long long)N * 32, T), T, 0, stream>>>(xC, xA, N, stats,
                                                             gn1_w, gn1_b, gn1_ms);

  // ---- GCN layer 2 ----
  k_gemm_k32<<<gemmBlocks, T, 0, stream>>>(xA, gcn2_w, xB, N);
  k_gcn_init<<<cdiv((long long)N * 32, T), T, 0, stream>>>(xB, deg, gcn2_b, xC, N);
  k_gcn_scatter<<<cdiv((long long)m * 32, T), T, 0, stream>>>(xB, src, dst, dinv, xC, m);
  k_zero<<<1, 64, 0, stream>>>(stats, 64);
  k_colstats<<<256, T, 0, stream>>>(xC, N, stats);
  k_gnorm_relu<<<cdiv((long long)N * 32, T), T, 0, stream>>>(xC, xA, N, stats,
                                                             gn2_w, gn2_b, gn2_ms);
  // xA now holds final node features

  // ---- edge MLPs (xe, mul) ----
  k_edge_mlp<<<cdiv(cdiv(m, 16), T / 32), T, 0, stream>>>(
      xA, src, dst, mlp1_w, mlp1_b, mlp2_w, mlp2_b, xe, mul, m);

  // ---- sparse-sparse bmm ----
  k_zero<<<cdiv(n_out * 32, T), T, 0, stream>>>(bmm, n_out * 32);
  k_bmm_scatter<<<cdiv(total * 32, T), T, 0, stream>>>(xe, mul, tri_a, tri_b,
                                                       tri_seg, bmm, total);

  // ---- mlp3 + GraphNorm + ReLU ----
  k_mlp3<<<cdiv(cdiv(n_out, 16), T / 32), T, 0, stream>>>(bmm, mlp3_w, is_edge,
                                                          mlp3_b, z, (int)n_out);
  k_zero<<<1, 64, 0, stream>>>(stats, 64);
  k_colstats<<<512, T, 0, stream>>>(z, (int)n_out, stats);
  k_gnorm_relu<<<cdiv(n_out * 32, T), T, 0, stream>>>(z, z, (int)n_out, stats,
                                                      gn3_w, gn3_b, gn3_ms);

  // ---- transpose product ----
  k_transmul<<<cdiv(n_out * 32, T), T, 0, stream>>>(z, trans_perm, y, n_out);

  // ---- final readout ----
  k_final<<<cdiv(P, T), T, 0, stream>>>(y, xA, pos, pred_idx, pred_mask, lind_w,
                                        lind_b, (float*)d_out, P);
}